// DifferentiateAttention_5806795784832
// MI455X (gfx1250) — compile-verified
//
#include <hip/hip_runtime.h>
#include <hip/hip_bf16.h>

// ---------------------------------------------------------------------------
// DifferentiateAttention for MI455X (gfx1250, wave32)
//   B=64 R=36 A=6 D=2048 DOUT=1024  -> BR=2304 rows, AP1=7 rows of v_P
// Bandwidth-bound (~158MB HBM @23.3TB/s ~ 7us floor). f32 WMMA 16x16x4 keeps
// exact f32 semantics; GEMM tiles staged by the Tensor Data Mover (TDM) with
// double-buffered LDS and s_wait_tensorcnt pipelining.
// ---------------------------------------------------------------------------

#define BR   2304      // B*R
#define AD   6         // A
#define AP1  7         // A+1 rows in v_P
#define DD   2048      // D
#define D2   4096      // 2*D
#define DOUT 1024
#define NPAIR 28       // upper-triangle incl. diagonal of 7x7
#define LDP  36        // padded LDS row stride (dwords): conflict-free frags

typedef __attribute__((ext_vector_type(2))) float v2f;
typedef __attribute__((ext_vector_type(8))) float v8f;
typedef unsigned int u32;
typedef __attribute__((ext_vector_type(4))) u32 u32x4;
typedef __attribute__((ext_vector_type(8))) int i32x8;
typedef __attribute__((ext_vector_type(4))) int i32x4;

// ---------------------------------------------------------------------------
// Kernel 0: g[d] = wx[d,d]*wx_bias[d]*wy[d,d]*wy_bias[d] / sqrt(D)
// ---------------------------------------------------------------------------
__global__ void g_kernel(const float* __restrict__ wx, const float* __restrict__ wy,
                         const float* __restrict__ wxb, const float* __restrict__ wyb,
                         float* __restrict__ g) {
    int d = blockIdx.x * blockDim.x + threadIdx.x;
    if (d < DD) {
        const float inv_sqrt_d = 0.022097086912079608f; // 1/sqrt(2048)
        g[d] = wx[(size_t)d * DD + d] * wxb[d] * wy[(size_t)d * DD + d] * wyb[d] * inv_sqrt_d;
    }
}

// ---------------------------------------------------------------------------
// Kernel 1: per (b,r) row -> concat[row][0:2048]=top, [2048:4096]=top-common
// One block of 256 threads (8 waves) per row; float4 streaming.
// ---------------------------------------------------------------------------
__global__ __launch_bounds__(256) void prep_kernel(
    const float* __restrict__ closest,  // (BR, A, D)
    const float* __restrict__ top,      // (BR, D)
    const float* __restrict__ g,        // (D,)
    float* __restrict__ ccat)           // (BR, 2D)
{
    __shared__ float sV[AP1 * DD];   // 7 x 2048 floats = 56KB
    __shared__ float sG[NPAIR];      // Gram sums (upper triangle)
    __shared__ float sC[AP1];        // per-row mean coefficients

    const int row  = blockIdx.x;
    const int tid  = threadIdx.x;
    const int lane = tid & 31;

    if (tid < NPAIR) sG[tid] = 0.0f;
    __syncthreads();

    const float* topRow = top + (size_t)row * DD;
    const float* clsRow = closest + (size_t)row * AD * DD;

    float acc[NPAIR];
#pragma unroll
    for (int p = 0; p < NPAIR; ++p) acc[p] = 0.0f;

    // Phase 1: stream rows once (float4), stash to LDS, accumulate 28 dots
#pragma unroll
    for (int it = 0; it < DD / (256 * 4); ++it) {
        int d4 = (tid + it * 256) * 4;
        float4 v[AP1];
        v[0] = *(const float4*)(topRow + d4);
#pragma unroll
        for (int a = 0; a < AD; ++a) v[a + 1] = *(const float4*)(clsRow + (size_t)a * DD + d4);
#pragma unroll
        for (int a = 0; a < AP1; ++a) *(float4*)(&sV[a * DD + d4]) = v[a];
        float4 gd = *(const float4*)(g + d4);
        int p = 0;
#pragma unroll
        for (int a = 0; a < AP1; ++a) {
            float4 vg;
            vg.x = v[a].x * gd.x; vg.y = v[a].y * gd.y;
            vg.z = v[a].z * gd.z; vg.w = v[a].w * gd.w;
#pragma unroll
            for (int e = a; e < AP1; ++e) {
                float s = acc[p];
                s = fmaf(vg.x, v[e].x, s);
                s = fmaf(vg.y, v[e].y, s);
                s = fmaf(vg.z, v[e].z, s);
                s = fmaf(vg.w, v[e].w, s);
                acc[p] = s;
                ++p;
            }
        }
    }

    // Phase 2: wave32 shuffle reduce, then LDS float atomics across 8 waves
#pragma unroll
    for (int p = 0; p < NPAIR; ++p) {
        float s = acc[p];
#pragma unroll
        for (int off = 16; off > 0; off >>= 1) s += __shfl_down(s, off);
        if (lane == 0) atomicAdd(&sG[p], s);
    }
    __syncthreads();

    // Phase 2b: 7-lane softmax; keep diagonal only, fold in 1/7 mean
    if (tid < AP1) {
        const int a = tid;
        float rowv[AP1];
#pragma unroll
        for (int e = 0; e < AP1; ++e) {
            int lo = a < e ? a : e;
            int hi = a < e ? e : a;
            int p = lo * AP1 - (lo * (lo - 1)) / 2 + (hi - lo);
            rowv[e] = sG[p];
        }
        float mx = rowv[0];
#pragma unroll
        for (int e = 1; e < AP1; ++e) mx = fmaxf(mx, rowv[e]);
        float sum = 0.0f;
#pragma unroll
        for (int e = 0; e < AP1; ++e) sum += __expf(rowv[e] - mx);
        sC[a] = (__expf(rowv[a] - mx) / sum) * (1.0f / 7.0f);
    }
    __syncthreads();

    // Phase 3: emit concat row (float4)
    float* outRow = ccat + (size_t)row * D2;
#pragma unroll
    for (int it = 0; it < DD / (256 * 4); ++it) {
        int d4 = (tid + it * 256) * 4;
        float4 common = make_float4(0.f, 0.f, 0.f, 0.f);
#pragma unroll
        for (int a = 0; a < AP1; ++a) {
            float c = sC[a];
            float4 va = *(const float4*)(&sV[a * DD + d4]);
            common.x = fmaf(c, va.x, common.x);
            common.y = fmaf(c, va.y, common.y);
            common.z = fmaf(c, va.z, common.z);
            common.w = fmaf(c, va.w, common.w);
        }
        float4 t = *(const float4*)(&sV[0 * DD + d4]);
        *(float4*)(outRow + d4) = t;
        float4 contr = make_float4(t.x - common.x, t.y - common.y,
                                   t.z - common.z, t.w - common.w);
        *(float4*)(outRow + DD + d4) = contr;
    }
}

// ---------------------------------------------------------------------------
// TDM 2D tile load: rows x (32 f32), LDS padded to 36-dword row stride.
// D# per CDNA5 ISA ch.8: group0 = {count/type/addr}, group1 = {dims/strides}.
//   pad_interval code 4 -> every 32 dwords; pad_amount code 3 -> 4 dwords.
// ---------------------------------------------------------------------------
#define TDM_G1W0 (0x00020000u /*data_size=4B*/ | (1u << 20) /*pad_enable*/ | \
                  (4u << 22) /*pad_interval=32dw*/ | (3u << 25) /*pad_amount=4dw*/)

__device__ __forceinline__ void tdm_load_2d(u32 lds_addr, const float* gptr,
                                            u32 tensor_d0, u32 tensor_d1,
                                            u32 tile_d0, u32 tile_d1) {
    unsigned long long ga = (unsigned long long)(uintptr_t)gptr;
    u32x4 g0;
    g0.x = 1u;                                   // count=1 (valid), user mode
    g0.y = lds_addr;                             // lds_addr [63:32]
    g0.z = (u32)ga;                              // global_addr [95:64]
    g0.w = (u32)((ga >> 32) & 0x01FFFFFFull)     // global_addr [120:96]
         | (2u << 30);                           // type=2 ("image") [127:126]
    i32x8 g1;
    g1[0] = (int)TDM_G1W0;                       // wg_mask=0, data_size, padding
    g1[1] = (int)(tensor_d0 << 16);              // tensor_dim0 lo16 @ [63:48]
    g1[2] = (int)((tensor_d0 >> 16) | (tensor_d1 << 16)); // dim0 hi / dim1 lo
    g1[3] = (int)((tensor_d1 >> 16) | (tile_d0 << 16));   // dim1 hi / tile_dim0
    g1[4] = (int)tile_d1;                        // tile_dim1; tile_dim2=0
    g1[5] = (int)tensor_d0;                      // tensor_dim0_stride lo32
    g1[6] = 0;                                   // stride0 hi / stride1 lo
    g1[7] = 0;                                   // stride1 hi
    i32x4 z4 = {0, 0, 0, 0};
    i32x8 z8 = {0, 0, 0, 0, 0, 0, 0, 0};
    __builtin_amdgcn_tensor_load_to_lds(g0, g1, z4, z4, z8, 0);
}

// ---------------------------------------------------------------------------
// Kernel 2: out(2304,1024) = relu(ccat(2304,4096) @ w^T(4096,1024) + bias)
// Block tile 128x64, 8 waves each 16x64. K chunks of 32 DMA'd by the TDM into
// double-buffered LDS; inner loop: v_wmma_f32_16x16x4_f32.
// ---------------------------------------------------------------------------
__global__ __launch_bounds__(256) void gemm_kernel(
    const float* __restrict__ Ccat,   // (BR, 4096)
    const float* __restrict__ W,      // (DOUT, 4096) row-major
    const float* __restrict__ bias,   // (DOUT,)
    float* __restrict__ out)          // (BR, DOUT)
{
    __shared__ float As[2][128 * LDP];   // 2 x 18432B
    __shared__ float Bs[2][64 * LDP];    // 2 x  9216B

    const int tid  = threadIdx.x;
    const int wave = tid >> 5;
    const int lane = tid & 31;
    const int half = lane >> 4;       // 0: K+0..1, 1: K+2..3 (A/B frag layout)
    const int lr   = lane & 15;

    const int m0 = blockIdx.x * 128;
    const int n0 = blockIdx.y * 64;

    // Scalar-branch wave-0 select so exec-ignoring TDM ops are truly skipped
    // by the other waves (s_cbranch on SGPR compare, not exec predication).
    const bool issuer = ((__builtin_amdgcn_readfirstlane(tid) >> 5) == 0);

    const float* Abase = Ccat + (size_t)m0 * D2;
    const float* Bbase = W + (size_t)n0 * D2;

    v8f acc[4];
#pragma unroll
    for (int j = 0; j < 4; ++j) acc[j] = (v8f){0.f,0.f,0.f,0.f,0.f,0.f,0.f,0.f};

    const int NC = D2 / 32;   // 128 K-chunks

    // Prologue: DMA chunk 0 into buffer 0
    if (issuer) {
        tdm_load_2d((u32)(uintptr_t)&As[0][0], Abase, D2, BR,   32, 128);
        tdm_load_2d((u32)(uintptr_t)&Bs[0][0], Bbase, D2, DOUT, 32, 64);
    }

    for (int c = 0; c < NC; ++c) {
        const int cur = c & 1;
        const int nxt = cur ^ 1;
        if (issuer) {
            if (c + 1 < NC) {
                const int kc = (c + 1) * 32;
                tdm_load_2d((u32)(uintptr_t)&As[nxt][0], Abase + kc, D2, BR,   32, 128);
                tdm_load_2d((u32)(uintptr_t)&Bs[nxt][0], Bbase + kc, D2, DOUT, 32, 64);
                __builtin_amdgcn_s_wait_tensorcnt(2);  // chunk c done (in-order)
            } else {
                __builtin_amdgcn_s_wait_tensorcnt(0);
            }
        }
        __syncthreads();   // chunk c visible to all waves

        const float* Ab = &As[cur][0];
        const float* Bb = &Bs[cur][0];
#pragma unroll
        for (int ks = 0; ks < 8; ++ks) {
            int k0 = ks * 4 + half * 2;
            // A frag: lanes 0-15 -> M rows @K k0..k0+1, lanes 16-31 @K+2..3
            v2f a = *(const v2f*)(&Ab[(wave * 16 + lr) * LDP + k0]);
#pragma unroll
            for (int j = 0; j < 4; ++j) {
                // B frag: B[k][n] = W[n][k]; same half-based K offset
                v2f b = *(const v2f*)(&Bb[(j * 16 + lr) * LDP + k0]);
                acc[j] = __builtin_amdgcn_wmma_f32_16x16x4_f32(
                    false, a, false, b, (short)0, acc[j], false, false);
            }
        }
        __syncthreads();   // all reads of `cur` done before TDM reuses it
    }

    // Epilogue: C/D layout -> VGPR i holds (M = i + 8*half, N = lr)
#pragma unroll
    for (int j = 0; j < 4; ++j) {
        int n = n0 + j * 16 + lr;
        float bv = bias[n];
#pragma unroll
        for (int i = 0; i < 8; ++i) {
            int m = m0 + wave * 16 + half * 8 + i;
            float v = acc[j][i] + bv;
            out[(size_t)m * DOUT + n] = v > 0.0f ? v : 0.0f;
        }
    }
}

// ---------------------------------------------------------------------------
extern "C" void kernel_launch(void* const* d_in, const int* in_sizes, int n_in,
                              void* d_out, int out_size, void* d_ws, size_t ws_size,
                              hipStream_t stream) {
    const float* closest = (const float*)d_in[0];
    const float* top     = (const float*)d_in[1];
    const float* wx      = (const float*)d_in[2];
    const float* wy      = (const float*)d_in[3];
    const float* wxb     = (const float*)d_in[4];
    const float* wyb     = (const float*)d_in[5];
    const float* w       = (const float*)d_in[6];
    const float* wb      = (const float*)d_in[7];
    float* out = (float*)d_out;

    float* g    = (float*)d_ws;                          // 2048 f32 (pad to 16KB)
    float* ccat = (float*)((char*)d_ws + 16384);         // 2304*4096 f32 = 37.75MB

    g_kernel<<<dim3((DD + 255) / 256), 256, 0, stream>>>(wx, wy, wxb, wyb, g);
    prep_kernel<<<dim3(BR), 256, 0, stream>>>(closest, top, g, ccat);
    gemm_kernel<<<dim3(BR / 128, DOUT / 64), 256, 0, stream>>>(ccat, w, wb, out);
}